// GlobalAttentionGeneral_41497974014299
// MI455X (gfx1250) — compile-verified
//
#include <hip/hip_runtime.h>
#include <hip/hip_bf16.h>

typedef __attribute__((ext_vector_type(2))) float v2f;
typedef __attribute__((ext_vector_type(8))) float v8f;

// Problem constants
#define NB    16
#define IDF   128
#define QL    16384          // 128*128
#define SRC   18
#define CDF   256
#define SDIM  100
#define CHW   (IDF * QL)     // 2097152 floats per batch of input
#define SP    132            // padded pixel stride in LDS (floats): 132 % 64 == 4 -> conflict-free b64 B-frags

// ---------------------------------------------------------------------------
// Kernel A: per-batch prep.
//   sourceT[b][i][s] = sum_c w_ctx[i][c] * context[b][c][s]      (128x18)
//   sent[b][i]       = sum_d w_lin[i][d] * sentence[b][d] + b_lin[i]
// ---------------------------------------------------------------------------
__global__ void __launch_bounds__(256) prep_kernel(
    const float* __restrict__ sentence, const float* __restrict__ context,
    const float* __restrict__ w_ctx, const float* __restrict__ w_lin,
    const float* __restrict__ b_lin, float* __restrict__ ws_st,
    float* __restrict__ ws_sent)
{
    const int b = blockIdx.x;
    const int tid = threadIdx.x;
    __shared__ float ctx[CDF * SRC];   // 4608 floats
    __shared__ float sen[SDIM];

    const float* ctxb = context + (size_t)b * CDF * SRC;
    for (int n = tid; n < CDF * SRC; n += 256) ctx[n] = ctxb[n];
    if (tid < SDIM) sen[tid] = sentence[b * SDIM + tid];
    __syncthreads();

    // sourceT: thread t -> row i = t>>1, 9 columns starting at (t&1)*9
    const int i  = tid >> 1;
    const int s0 = (tid & 1) * 9;
    float acc[9];
#pragma unroll
    for (int j = 0; j < 9; ++j) acc[j] = 0.f;
    const float* wrow = w_ctx + i * CDF;
    for (int c = 0; c < CDF; ++c) {
        const float w = wrow[c];
#pragma unroll
        for (int j = 0; j < 9; ++j) acc[j] = fmaf(w, ctx[c * SRC + s0 + j], acc[j]);
    }
    float* stb = ws_st + b * (IDF * SRC) + i * SRC + s0;
#pragma unroll
    for (int j = 0; j < 9; ++j) stb[j] = acc[j];

    // sent (linear layer), threads 0..127
    if (tid < IDF) {
        float a = b_lin[tid];
        const float* wl = w_lin + tid * SDIM;
        for (int d = 0; d < SDIM; ++d) a = fmaf(wl[d], sen[d], a);
        ws_sent[b * IDF + tid] = a;
    }
}

// ---------------------------------------------------------------------------
// Kernel B: attention path (memory-bound -> VALU).
// One thread per pixel q:
//   logits[s] = sum_i input[b,i,q] * sourceT[b,i,s]     (18 accumulators)
//   p = softmax(logits)           -> word_attn[b,s,q]
//   wC[b,i,q] = sum_s sourceT[b,i,s] * p[s]
// sourceT read via uniform pointer -> scalar loads.
// ---------------------------------------------------------------------------
__global__ void __launch_bounds__(256) attn_kernel(
    const float* __restrict__ input, const float* __restrict__ ws_st,
    float* __restrict__ out_wc, float* __restrict__ out_wa)
{
    const int b = blockIdx.y;
    const int q = blockIdx.x * 256 + threadIdx.x;
    const float* stb = ws_st + b * (IDF * SRC);       // wave-uniform
    const float* inb = input + (size_t)b * CHW + q;

    float acc[SRC];
#pragma unroll
    for (int s = 0; s < SRC; ++s) acc[s] = 0.f;

    for (int i = 0; i < IDF; ++i) {
        const float x = inb[(size_t)i * QL];          // coalesced
#pragma unroll
        for (int s = 0; s < SRC; ++s) acc[s] = fmaf(x, stb[i * SRC + s], acc[s]);
    }

    // softmax over 18
    float m = acc[0];
#pragma unroll
    for (int s = 1; s < SRC; ++s) m = fmaxf(m, acc[s]);
    float sum = 0.f;
#pragma unroll
    for (int s = 0; s < SRC; ++s) { acc[s] = __expf(acc[s] - m); sum += acc[s]; }
    const float r = 1.f / sum;

    float* wa = out_wa + (size_t)b * (SRC * QL) + q;
#pragma unroll
    for (int s = 0; s < SRC; ++s) {
        acc[s] *= r;
        __builtin_nontemporal_store(acc[s], wa + (size_t)s * QL);
    }

    float* wc = out_wc + (size_t)b * CHW + q;
    for (int i = 0; i < IDF; ++i) {
        float v = 0.f;
#pragma unroll
        for (int s = 0; s < SRC; ++s) v = fmaf(stb[i * SRC + s], acc[s], v);
        __builtin_nontemporal_store(v, wc + (size_t)i * QL);
    }
}

// ---------------------------------------------------------------------------
// Kernel C: sv = w_sv @ (input * sent) with fp32 WMMA, fused channel softmax.
// Block = (batch, 64-pixel tile). 8 waves; wave w owns channels [16w,16w+16).
// LDS x[p][k] (pad stride 132) holds input*sent; B-frags via ds_load_b64.
// A-frags (w_sv stripe) preloaded to registers once (32 x v2f per wave).
// 32 k-steps x 4 pixel subtiles of v_wmma_f32_16x16x4_f32 per wave.
// ---------------------------------------------------------------------------
__global__ void __launch_bounds__(256) sv_kernel(
    const float* __restrict__ input, const float* __restrict__ w_sv,
    const float* __restrict__ ws_sent, float* __restrict__ out_wS,
    float* __restrict__ out_sa)
{
    const int b   = blockIdx.y;
    const int q0  = blockIdx.x * 64;
    const int tid = threadIdx.x;

    __shared__ float xs[64 * SP];   // 8448 floats; reused as sv[128][66] after GEMM
    __shared__ float sents[IDF];
    __shared__ float red[256];
    __shared__ float pmax[64];
    __shared__ float rsum[64];

    if (tid < IDF) sents[tid] = ws_sent[b * IDF + tid];
    __syncthreads();

    // Stage x[p][ch] = input[b,ch,q0+p] * sent[ch], transposed + padded.
    const float* inb = input + (size_t)b * CHW + q0;
#pragma unroll
    for (int j = 0; j < 32; ++j) {
        const int n = tid + 256 * j;
        const int ch = n >> 6, p = n & 63;
        xs[p * SP + ch] = inb[(size_t)ch * QL + p] * sents[ch];
    }
    __syncthreads();

    const int lane = tid & 31;
    const int w    = tid >> 5;       // wave id 0..7 -> channel stripe
    const int half = lane >> 4;      // K-pair select (ISA 16x4 f32 layout)
    const int ln   = lane & 15;
    const int chb  = w * 16;

    // Preload A fragments: w_sv[chb+ln][k + 2*half .. +1], k = 4*kk
    v2f af[32];
    const float* wrow = w_sv + (size_t)(chb + ln) * IDF + 2 * half;
#pragma unroll
    for (int kk = 0; kk < 32; ++kk)
        af[kk] = *reinterpret_cast<const v2f*>(wrow + 4 * kk);

    v8f acc[4];
#pragma unroll
    for (int t = 0; t < 4; ++t) {
        v8f a = {0.f, 0.f, 0.f, 0.f, 0.f, 0.f, 0.f, 0.f};
        const int pl = t * 16 + ln;                      // B: N = lane (mod 16)
        const float* bp = &xs[pl * SP + 2 * half];
#pragma unroll
        for (int kk = 0; kk < 32; ++kk) {
            const v2f bv = *reinterpret_cast<const v2f*>(bp + 4 * kk);  // ds_load_b64
            a = __builtin_amdgcn_wmma_f32_16x16x4_f32(
                    false, af[kk], false, bv, (short)0, a, false, false);
        }
        acc[t] = a;
    }
    __syncthreads();   // all waves done reading xs -> safe to overwrite

    // Scatter D fragments into xs reinterpreted as sv[ch][66]:
    // VGPR r: lanes 0-15 -> M=r, lanes 16-31 -> M=r+8 (ISA C/D layout)
#pragma unroll
    for (int t = 0; t < 4; ++t) {
        const int p = t * 16 + ln;
#pragma unroll
        for (int r = 0; r < 8; ++r) {
            const int ch = chb + r + half * 8;
            xs[ch * 66 + p] = acc[t][r];
        }
    }
    __syncthreads();

    // Channel softmax per pixel: 4 threads/pixel, 32 channels each.
    const int p  = tid & 63;
    const int qt = tid >> 6;
    float m = -3.402823466e38f;
    for (int c = 0; c < 32; ++c) m = fmaxf(m, xs[(qt * 32 + c) * 66 + p]);
    red[tid] = m;
    __syncthreads();
    if (tid < 64)
        pmax[tid] = fmaxf(fmaxf(red[tid], red[tid + 64]),
                          fmaxf(red[tid + 128], red[tid + 192]));
    __syncthreads();
    const float mm = pmax[p];
    float s = 0.f;
    for (int c = 0; c < 32; ++c) s += __expf(xs[(qt * 32 + c) * 66 + p] - mm);
    red[tid] = s;
    __syncthreads();
    if (tid < 64)
        rsum[tid] = 1.0f / (red[tid] + red[tid + 64] + red[tid + 128] + red[tid + 192]);
    __syncthreads();

    // Outputs: sent_att and weightedSentence (coalesced, non-temporal).
    float* sa = out_sa + (size_t)b * CHW + q0;
    float* wS = out_wS + (size_t)b * CHW + q0;
#pragma unroll
    for (int j = 0; j < 32; ++j) {
        const int n = tid + 256 * j;
        const int ch = n >> 6, pp = n & 63;
        const float a = __expf(xs[ch * 66 + pp] - pmax[pp]) * rsum[pp];
        __builtin_nontemporal_store(a, sa + (size_t)ch * QL + pp);
        __builtin_nontemporal_store(sents[ch] * a, wS + (size_t)ch * QL + pp);
    }
}

// ---------------------------------------------------------------------------
extern "C" void kernel_launch(void* const* d_in, const int* in_sizes, int n_in,
                              void* d_out, int out_size, void* d_ws, size_t ws_size,
                              hipStream_t stream) {
    const float* input    = (const float*)d_in[0];
    const float* sentence = (const float*)d_in[1];
    const float* context  = (const float*)d_in[2];
    const float* w_ctx    = (const float*)d_in[3];
    const float* w_lin    = (const float*)d_in[4];
    const float* b_lin    = (const float*)d_in[5];
    const float* w_sv     = (const float*)d_in[6];

    float* out = (float*)d_out;
    float* o_wc = out;                         // weightedContext : 33554432
    float* o_wS = out + (size_t)NB * CHW;      // weightedSentence: 33554432
    float* o_wa = o_wS + (size_t)NB * CHW;     // word_attn       :  4718592
    float* o_sa = o_wa + (size_t)NB * SRC * QL;// sent_att        : 33554432

    float* ws_st   = (float*)d_ws;             // 16*128*18 = 36864 floats
    float* ws_sent = ws_st + NB * IDF * SRC;   // 16*128    =  2048 floats

    prep_kernel<<<NB, 256, 0, stream>>>(sentence, context, w_ctx, w_lin, b_lin,
                                        ws_st, ws_sent);
    attn_kernel<<<dim3(QL / 256, NB), 256, 0, stream>>>(input, ws_st, o_wc, o_wa);
    sv_kernel<<<dim3(QL / 64, NB), 256, 0, stream>>>(input, w_sv, ws_sent, o_wS, o_sa);
}